// Method_GCN_48756468744279
// MI455X (gfx1250) — compile-verified
//
#include <hip/hip_runtime.h>
#include <math.h>

typedef __attribute__((ext_vector_type(2))) float v2f;
typedef __attribute__((ext_vector_type(8))) float v8f;

// ---------------------------------------------------------------------------
// Elementwise / degree kernels
// ---------------------------------------------------------------------------
__global__ void k_zero(float* __restrict__ p, int n) {
    int i = blockIdx.x * blockDim.x + threadIdx.x;
    if (i < n) p[i] = 0.0f;
}

__global__ void k_count(const int* __restrict__ dst, float* __restrict__ deg, int E) {
    int e = blockIdx.x * blockDim.x + threadIdx.x;
    if (e < E) unsafeAtomicAdd(&deg[dst[e]], 1.0f);
}

__global__ void k_rsqrt(float* __restrict__ dis, int n) {
    int i = blockIdx.x * blockDim.x + threadIdx.x;
    if (i < n) dis[i] = rsqrtf(dis[i] + 1.0f);
}

// agg[i][f] = dis[i]^2 * h[i][f]   (self-loop term), F = feature count
template <int F>
__global__ void k_selfinit(const float* __restrict__ h, const float* __restrict__ dis,
                           float* __restrict__ agg, int total) {
    int i = blockIdx.x * blockDim.x + threadIdx.x;
    if (i >= total) return;
    int r = i / F;
    float d = dis[r];
    agg[i] = d * d * h[i];
}

// agg[dst][f] += dis[src]*dis[dst]*h[src][f]
template <int F>
__global__ void k_scatter(const int* __restrict__ src, const int* __restrict__ dst,
                          const float* __restrict__ dis, const float* __restrict__ h,
                          float* __restrict__ agg, int E) {
    int t = blockIdx.x * blockDim.x + threadIdx.x;
    if (t >= E * F) return;
    int e = t / F;
    int f = t - e * F;
    int s = src[e];
    int d = dst[e];
    float norm = dis[s] * dis[d];
    unsafeAtomicAdd(&agg[(size_t)d * F + f], norm * h[(size_t)s * F + f]);
}

// in-place h = relu(h + b)
template <int F>
__global__ void k_bias_relu(float* __restrict__ h, const float* __restrict__ b, int total) {
    int i = blockIdx.x * blockDim.x + threadIdx.x;
    if (i >= total) return;
    int f = i & (F - 1);
    h[i] = fmaxf(h[i] + b[f], 0.0f);
}

// per-row: out = (row + b2) - logsumexp(row + b2), C = 16
__global__ void k_log_softmax16(float* __restrict__ out, const float* __restrict__ b2, int N) {
    int i = blockIdx.x * blockDim.x + threadIdx.x;
    if (i >= N) return;
    float* row = out + (size_t)i * 16;
    float v[16];
    float m = -INFINITY;
#pragma unroll
    for (int c = 0; c < 16; ++c) {
        v[c] = row[c] + b2[c];
        m = fmaxf(m, v[c]);
    }
    float s = 0.0f;
#pragma unroll
    for (int c = 0; c < 16; ++c) s += expf(v[c] - m);
    float ls = m + logf(s);
#pragma unroll
    for (int c = 0; c < 16; ++c) row[c] = v[c] - ls;
}

// ---------------------------------------------------------------------------
// GEMM1: H[N][64] = X[N][512] @ W[512][64]  (f32 WMMA 16x16x4)
// One wave computes a 16x64 output stripe (4 accumulators).
// Block = 256 threads = 8 waves = 128 rows. W^T staged in 32KB LDS K-chunks.
// ---------------------------------------------------------------------------
__global__ __launch_bounds__(256) void k_gemm1(const float* __restrict__ X,
                                               const float* __restrict__ W,
                                               float* __restrict__ H, int N) {
    __shared__ float wt[64 * 128];  // [n][k_local], k-chunk of 128

    const int tid = threadIdx.x;
    const int lane = tid & 31;
    const int wave = tid >> 5;
    const int rowBase = blockIdx.x * 128 + wave * 16;
    const int rl = lane & 15;            // lane row (A) / lane col (B)
    const int kh = (lane >> 4) << 1;     // K sub-offset: 0 or 2
    const int arow = min(rowBase + rl, N - 1);
    const float* aBase = X + (size_t)arow * 512 + kh;

    v8f acc0 = {}, acc1 = {}, acc2 = {}, acc3 = {};

    for (int ko = 0; ko < 4; ++ko) {
        __syncthreads();
        // Stage W[ko*128 .. ko*128+128)[0..64) transposed into LDS
        for (int idx = tid; idx < 64 * 128; idx += 256) {
            int kl = idx >> 6;
            int n = idx & 63;
            wt[n * 128 + kl] = W[(size_t)(ko * 128 + kl) * 64 + n];
        }
        __syncthreads();

        const float* a0 = aBase + ko * 128;
        const float* bc = wt + kh;
#pragma unroll 8
        for (int kc = 0; kc < 32; ++kc) {
            v2f a = *(const v2f*)(a0 + kc * 4);
            const float* b0 = bc + kc * 4;
            v2f bf0 = *(const v2f*)(b0 + (size_t)(rl + 0) * 128);
            v2f bf1 = *(const v2f*)(b0 + (size_t)(rl + 16) * 128);
            v2f bf2 = *(const v2f*)(b0 + (size_t)(rl + 32) * 128);
            v2f bf3 = *(const v2f*)(b0 + (size_t)(rl + 48) * 128);
            acc0 = __builtin_amdgcn_wmma_f32_16x16x4_f32(false, a, false, bf0, (short)0, acc0, false, false);
            acc1 = __builtin_amdgcn_wmma_f32_16x16x4_f32(false, a, false, bf1, (short)0, acc1, false, false);
            acc2 = __builtin_amdgcn_wmma_f32_16x16x4_f32(false, a, false, bf2, (short)0, acc2, false, false);
            acc3 = __builtin_amdgcn_wmma_f32_16x16x4_f32(false, a, false, bf3, (short)0, acc3, false, false);
        }
    }

    // C/D layout: VGPR i -> row rowBase+i (lanes 0-15) / rowBase+i+8 (lanes 16-31)
    const int srow = rowBase + (lane >> 4) * 8;
#pragma unroll
    for (int i = 0; i < 8; ++i) {
        int r = srow + i;
        if (r < N) {
            float* o = H + (size_t)r * 64 + rl;
            o[0]  = acc0[i];
            o[16] = acc1[i];
            o[32] = acc2[i];
            o[48] = acc3[i];
        }
    }
}

// ---------------------------------------------------------------------------
// GEMM2: H2[N][16] = H1[N][64] @ W2[64][16]  (f32 WMMA 16x16x4)
// One wave computes a 16x16 output tile; W2^T (4KB) fully in LDS.
// ---------------------------------------------------------------------------
__global__ __launch_bounds__(256) void k_gemm2(const float* __restrict__ H1,
                                               const float* __restrict__ W2,
                                               float* __restrict__ H2, int N) {
    __shared__ float wt[16 * 64];  // [n][k]

    const int tid = threadIdx.x;
    for (int idx = tid; idx < 16 * 64; idx += 256) {
        int k = idx >> 4;
        int n = idx & 15;
        wt[n * 64 + k] = W2[idx];
    }
    __syncthreads();

    const int lane = tid & 31;
    const int wave = tid >> 5;
    const int rowBase = blockIdx.x * 128 + wave * 16;
    const int rl = lane & 15;
    const int kh = (lane >> 4) << 1;
    const int arow = min(rowBase + rl, N - 1);
    const float* aBase = H1 + (size_t)arow * 64 + kh;
    const float* bBase = wt + rl * 64 + kh;

    v8f acc = {};
#pragma unroll
    for (int kc = 0; kc < 16; ++kc) {
        v2f a = *(const v2f*)(aBase + kc * 4);
        v2f b = *(const v2f*)(bBase + kc * 4);
        acc = __builtin_amdgcn_wmma_f32_16x16x4_f32(false, a, false, b, (short)0, acc, false, false);
    }

    const int srow = rowBase + (lane >> 4) * 8;
#pragma unroll
    for (int i = 0; i < 8; ++i) {
        int r = srow + i;
        if (r < N) H2[(size_t)r * 16 + rl] = acc[i];
    }
}

// ---------------------------------------------------------------------------
// Launch
// ---------------------------------------------------------------------------
extern "C" void kernel_launch(void* const* d_in, const int* in_sizes, int n_in,
                              void* d_out, int out_size, void* d_ws, size_t ws_size,
                              hipStream_t stream) {
    const float* x  = (const float*)d_in[0];
    const int*  edge = (const int*)d_in[1];
    const float* W1 = (const float*)d_in[2];
    const float* b1 = (const float*)d_in[3];
    const float* W2 = (const float*)d_in[4];
    const float* b2 = (const float*)d_in[5];

    const int N = in_sizes[0] / 512;
    const int E = in_sizes[1] / 2;
    const int* src = edge;
    const int* dst = edge + E;

    float* out = (float*)d_out;              // N x 16: agg2 accumulator -> final
    float* ws = (float*)d_ws;
    float* dis  = ws;                        // N
    float* bufA = ws + (((size_t)N + 255) & ~(size_t)255);  // N x 64: h1pre, then h2pre
    float* bufB = bufA + (size_t)N * 64;     // N x 64: agg1 -> h1

    const int T = 256;
    auto blocks = [](long long total, int thr) { return (unsigned)((total + thr - 1) / thr); };

    // degrees -> d^{-1/2}
    k_zero<<<blocks(N, T), T, 0, stream>>>(dis, N);
    k_count<<<blocks(E, T), T, 0, stream>>>(dst, dis, E);
    k_rsqrt<<<blocks(N, T), T, 0, stream>>>(dis, N);

    // layer 1: h1pre = x @ W1
    k_gemm1<<<blocks((long long)N, 128), T, 0, stream>>>(x, W1, bufA, N);
    // agg1 = dis^2 * h1pre  (self loops)
    k_selfinit<64><<<blocks((long long)N * 64, T), T, 0, stream>>>(bufA, dis, bufB, N * 64);
    // agg1 += scatter over edges
    k_scatter<64><<<blocks((long long)E * 64, T), T, 0, stream>>>(src, dst, dis, bufA, bufB, E);
    // h1 = relu(agg1 + b1)   (in place in bufB)
    k_bias_relu<64><<<blocks((long long)N * 64, T), T, 0, stream>>>(bufB, b1, N * 64);

    // layer 2: h2pre = h1 @ W2   (reuse bufA, only N*16 needed)
    k_gemm2<<<blocks((long long)N, 128), T, 0, stream>>>(bufB, W2, bufA, N);
    // agg2 (in d_out) = dis^2 * h2pre
    k_selfinit<16><<<blocks((long long)N * 16, T), T, 0, stream>>>(bufA, dis, out, N * 16);
    // agg2 += scatter over edges
    k_scatter<16><<<blocks((long long)E * 16, T), T, 0, stream>>>(src, dst, dis, bufA, out, E);
    // out = log_softmax(agg2 + b2)
    k_log_softmax16<<<blocks(N, T), T, 0, stream>>>(out, b2, N);
}